// LVC_ConvBlock_56968446214146
// MI455X (gfx1250) — compile-verified
//
#include <hip/hip_runtime.h>
#include <hip/hip_bf16.h>

typedef __attribute__((ext_vector_type(2))) float v2f;
typedef __attribute__((ext_vector_type(8))) float v8f;

#define B_ 8
#define C_ 128
#define T_ 32768
#define L_ 256
#define K_ 3
#define HOP_ 128
#define SLOPE_ 0.2f

// Transposed LDS tile stride over the channel dimension (128 cols padded to
// 132: 132 % 64 == 4, so a ds_load_b64 across 32 lanes touches all 64 banks).
#define TS 132

__device__ __forceinline__ float lrelu(float v) {
    return v > 0.0f ? v : SLOPE_ * v;
}

// -------------------------------------------------------------------------
// Kernel 0: per-output-channel weight-norm scale: scale[o] = g[o]/||v[o,:,:]||
// -------------------------------------------------------------------------
__global__ void wn_scale_kernel(const float* __restrict__ conv_v,
                                const float* __restrict__ conv_g,
                                float* __restrict__ scale) {
    int o = threadIdx.x;
    if (o < C_) {
        float s = 0.0f;
        for (int j = 0; j < C_ * K_; ++j) {
            float v = conv_v[o * (C_ * K_) + j];
            s += v * v;
        }
        scale[o] = conv_g[o] * rsqrtf(s);
    }
}

// -------------------------------------------------------------------------
// Kernel 1: location-variable conv.  One block per (b, s): 128x128 GEMM,
// reduction over (K taps x Cin) = 384 in fp32 via v_wmma_f32_16x16x4_f32.
// Reduction ordered tap-major (r = k*128 + i) and the x tile staged
// TRANSPOSED (xsT[t][i]) so each B fragment is one aligned ds_load_b64.
// -------------------------------------------------------------------------
__global__ void __launch_bounds__(256)
lvc_kernel(const float* __restrict__ x,
           const float* __restrict__ weight,
           const float* __restrict__ bias,
           float* __restrict__ h) {
    extern __shared__ float xsT[];  // [HOP_+2][TS]  (t-major, channel inner)
    const int s = blockIdx.x;
    const int b = blockIdx.y;
    const int tid = threadIdx.x;
    const int lane = tid & 31;
    const int wave = tid >> 5;          // 8 waves, each owns 16 output rows
    const int o0 = wave * 16;
    const int half = lane >> 4;         // 0: lanes 0-15, 1: lanes 16-31
    const int lx = lane & 15;

    // Stage xp[b, :, s*128-1 .. s*128+129] transposed into LDS
    for (int idx = tid; idx < C_ * (HOP_ + 2); idx += 256) {
        int i = idx / (HOP_ + 2);       // channel
        int j = idx - i * (HOP_ + 2);   // padded t position
        int g = s * HOP_ + j - 1;
        float v = 0.0f;
        if (g >= 0 && g < T_) v = x[(b * C_ + i) * T_ + g];
        xsT[j * TS + i] = v;
    }
    __syncthreads();

    v8f acc[8];
    #pragma unroll
    for (int n = 0; n < 8; ++n) acc[n] = (v8f)(0.0f);

    const int row = o0 + lx;  // A-matrix M for this lane
    #pragma unroll 1
    for (int k = 0; k < K_; ++k) {
        #pragma unroll 1
        for (int i0 = 0; i0 < C_; i0 += 4) {
            const int ik = i0 + half * 2;   // even -> 8B-aligned LDS pairs

            v2f a;
            a.x = weight[(((size_t)(b * C_ + ik)     * C_ + row) * K_ + k) * L_ + s];
            a.y = weight[(((size_t)(b * C_ + ik + 1) * C_ + row) * K_ + k) * L_ + s];

            #pragma unroll
            for (int n = 0; n < 8; ++n) {
                const int col = n * 16 + lx;
                const v2f bb = *(const v2f*)(xsT + (col + k) * TS + ik);
                acc[n] = __builtin_amdgcn_wmma_f32_16x16x4_f32(
                    false, a, false, bb, (short)0, acc[n], false, false);
            }
        }
    }

    // Epilogue: bias + leaky relu, write h (B,C,T)
    #pragma unroll
    for (int n = 0; n < 8; ++n) {
        const int col = n * 16 + lx;
        #pragma unroll
        for (int v = 0; v < 8; ++v) {
            const int m = o0 + v + half * 8;
            float val = acc[n][v] + bias[(b * C_ + m) * L_ + s];
            h[(size_t)(b * C_ + m) * T_ + s * HOP_ + col] = lrelu(val);
        }
    }
}

// -------------------------------------------------------------------------
// Kernel 2: dilated (d=3) conv over channels.  One block per (b, 128-wide t
// tile).  3 taps x (Cin=128) reduction via wmma f32 16x16x4, h tile staged
// transposed with halo 3; tap shift is a row offset in the transposed tile.
// Row scale (weight norm) + conv bias + leaky relu in epilogue.
// -------------------------------------------------------------------------
__global__ void __launch_bounds__(256)
dconv_kernel(const float* __restrict__ h,
             const float* __restrict__ conv_v,
             const float* __restrict__ scale,
             const float* __restrict__ conv_b,
             float* __restrict__ out) {
    extern __shared__ float hsT[];  // [134][TS]
    const int tile = blockIdx.x;
    const int b = blockIdx.y;
    const int t0 = tile * 128;
    const int tid = threadIdx.x;
    const int lane = tid & 31;
    const int wave = tid >> 5;
    const int o0 = wave * 16;
    const int half = lane >> 4;
    const int lx = lane & 15;

    // Stage h[b, :, t0-3 .. t0+130] transposed into LDS (halo 3 each side)
    for (int idx = tid; idx < C_ * 134; idx += 256) {
        int i = idx / 134;
        int j = idx - i * 134;
        int g = t0 + j - 3;
        float v = 0.0f;
        if (g >= 0 && g < T_) v = h[(size_t)(b * C_ + i) * T_ + g];
        hsT[j * TS + i] = v;
    }
    __syncthreads();

    v8f acc[8];
    #pragma unroll
    for (int n = 0; n < 8; ++n) acc[n] = (v8f)(0.0f);

    const int row = o0 + lx;
    #pragma unroll 1
    for (int d = 0; d < 3; ++d) {
        const int shift = d * 3;  // t_local + 3 + (d-1)*3
        #pragma unroll 1
        for (int i0 = 0; i0 < C_; i0 += 4) {
            const int ik = i0 + half * 2;

            v2f a;
            a.x = conv_v[(row * C_ + ik)     * K_ + d];
            a.y = conv_v[(row * C_ + ik + 1) * K_ + d];

            #pragma unroll
            for (int n = 0; n < 8; ++n) {
                const int col = n * 16 + lx;
                const v2f bb = *(const v2f*)(hsT + (col + shift) * TS + ik);
                acc[n] = __builtin_amdgcn_wmma_f32_16x16x4_f32(
                    false, a, false, bb, (short)0, acc[n], false, false);
            }
        }
    }

    // Epilogue: row scale (weight norm), bias, leaky relu
    #pragma unroll
    for (int n = 0; n < 8; ++n) {
        const int col = n * 16 + lx;
        #pragma unroll
        for (int v = 0; v < 8; ++v) {
            const int m = o0 + v + half * 8;
            float val = acc[n][v] * scale[m] + conv_b[m];
            out[(size_t)(b * C_ + m) * T_ + t0 + col] = lrelu(val);
        }
    }
}

// -------------------------------------------------------------------------
extern "C" void kernel_launch(void* const* d_in, const int* in_sizes, int n_in,
                              void* d_out, int out_size, void* d_ws, size_t ws_size,
                              hipStream_t stream) {
    const float* x      = (const float*)d_in[0];
    const float* weight = (const float*)d_in[1];
    const float* bias   = (const float*)d_in[2];
    const float* conv_v = (const float*)d_in[3];
    const float* conv_g = (const float*)d_in[4];
    const float* conv_b = (const float*)d_in[5];
    float* out = (float*)d_out;

    float* scale = (float*)d_ws;                       // 128 floats
    float* h     = (float*)((char*)d_ws + 1024);       // B*C*T floats

    wn_scale_kernel<<<1, 128, 0, stream>>>(conv_v, conv_g, scale);

    dim3 grid(L_, B_);
    size_t lds1 = (size_t)(HOP_ + 2) * TS * sizeof(float);  // ~68.6 KB
    lvc_kernel<<<grid, 256, lds1, stream>>>(x, weight, bias, h);

    size_t lds2 = (size_t)134 * TS * sizeof(float);         // ~70.8 KB
    dconv_kernel<<<grid, 256, lds2, stream>>>(h, conv_v, scale, conv_b, out);
}